// RoIHeadsOur_55894704390549
// MI455X (gfx1250) — compile-verified
//
#include <hip/hip_runtime.h>
#include <hip/hip_bf16.h>
#include <climits>

// ---------------- problem constants ----------------
#define B_IMG   4
#define N_PROP  4096
#define NCLS    91
#define CM1     90            // classes without background
#define FEAT    1024
#define CAND    (N_PROP * CM1)   // 368640 candidates per image
#define IMG_W_F 1216.0f
#define IMG_H_F 800.0f
#define BBOX_CLIP 4.135166556742356f   // log(1000/16)
#define DETS    100

typedef float v2f  __attribute__((ext_vector_type(2)));
typedef float v8f  __attribute__((ext_vector_type(8)));

// ---------------- GEMM: C[M,N] = A[M,1024] * W[1024,N] + bias ----------------
// Block = 256 threads = 8 waves (wave32). Block tile = 256(M) x 32(N).
// Each wave computes a 32x32 output (2x2 grid of 16x16 WMMA accumulators).
// B is staged TRANSPOSED in LDS so every A/B fragment is one aligned ds b64.
#define MT 256
#define NT 32
#define KC 32
#define AP (KC + 4)   // A row stride 36: 16B-aligned stores, conflict-free frag reads
#define BTP (KC + 2)  // Bt row stride 34: 8B-aligned b64 frags, conflict-free rows

__global__ __launch_bounds__(256) void gemm_bias_wmma(
    const float* __restrict__ A, const float* __restrict__ W,
    const float* __restrict__ bias, float* __restrict__ Cout, int N)
{
  __shared__ float As[MT][AP];    // 36 KB
  __shared__ float Bt[NT][BTP];   // Bt[n][k], ~4.25 KB

  const int tid  = threadIdx.x;
  const int wave = tid >> 5;
  const int lane = tid & 31;
  const int nl   = lane & 15;   // N-lane (also M-lane for A frags)
  const int kh   = lane >> 4;   // half-wave K selector
  const int m0   = blockIdx.x * MT;
  const int n0   = blockIdx.y * NT;
  const int wm   = wave << 5;   // wave's 32-row M slice

  v8f a00 = {}, a01 = {}, a10 = {}, a11 = {};

  for (int k0 = 0; k0 < FEAT; k0 += KC) {
    // A chunk 256x32, float4 coalesced: 8192 floats / 256 thr = 8 float4 each
    {
      const int r  = tid >> 3;          // 0..31
      const int c4 = (tid & 7) << 2;    // 0..28
#pragma unroll
      for (int rr = 0; rr < MT; rr += 32) {
        const float4 v = *(const float4*)(A + (size_t)(m0 + r + rr) * FEAT + k0 + c4);
        *(float4*)(&As[r + rr][c4]) = v;
      }
    }
    // B chunk 32(K)x32(N) -> transposed into Bt[n][k]; global reads coalesced in n
    {
#pragma unroll
      for (int i = 0; i < (KC * NT) / 256; ++i) {
        const int idx = tid + i * 256;
        const int r = idx >> 5, c = idx & 31;   // r = k row, c = n col
        const int n = n0 + c;
        Bt[c][r] = (n < N) ? W[(size_t)(k0 + r) * N + n] : 0.0f;
      }
    }
    __syncthreads();

#pragma unroll
    for (int kk = 0; kk < KC; kk += 4) {
      // A 16x4 f32 frags: lane=M, VGPR0=K{0|2}, VGPR1=K{1|3} by half-wave.
      // B 4x16 frags mirror A; Bt makes each one a contiguous aligned b64.
      const int ks = kk + 2 * kh;
      v2f a0 = *(const v2f*)(&As[wm + nl][ks]);
      v2f a1 = *(const v2f*)(&As[wm + 16 + nl][ks]);
      v2f b0 = *(const v2f*)(&Bt[nl][ks]);
      v2f b1 = *(const v2f*)(&Bt[16 + nl][ks]);
      a00 = __builtin_amdgcn_wmma_f32_16x16x4_f32(false, a0, false, b0, (short)0, a00, false, false);
      a01 = __builtin_amdgcn_wmma_f32_16x16x4_f32(false, a0, false, b1, (short)0, a01, false, false);
      a10 = __builtin_amdgcn_wmma_f32_16x16x4_f32(false, a1, false, b0, (short)0, a10, false, false);
      a11 = __builtin_amdgcn_wmma_f32_16x16x4_f32(false, a1, false, b1, (short)0, a11, false, false);
    }
    __syncthreads();
  }

  // Epilogue: C/D layout -> VGPR r holds M = r (lanes 0-15) / r+8 (lanes 16-31)
  {
    const int nA = n0 + nl;
    const int nB = n0 + 16 + nl;
    const float bvA = (nA < N) ? bias[nA] : 0.0f;
    const float bvB = (nB < N) ? bias[nB] : 0.0f;
#pragma unroll
    for (int r = 0; r < 8; ++r) {
      const int mA = m0 + wm + r + 8 * kh;
      const int mB = mA + 16;
      if (nA < N) {
        Cout[(size_t)mA * N + nA] = a00[r] + bvA;
        Cout[(size_t)mB * N + nA] = a10[r] + bvA;
      }
      if (nB < N) {
        Cout[(size_t)mA * N + nB] = a01[r] + bvB;
        Cout[(size_t)mB * N + nB] = a11[r] + bvB;
      }
    }
  }
}

// ---------------- zero per-image candidate counters ----------------
__global__ void zero_cnt_kernel(int* c) {
  if (threadIdx.x < B_IMG) c[threadIdx.x] = 0;
}

// ---------------- softmax + decode + clip + compact ----------------
// One wave per proposal row.
__global__ __launch_bounds__(256) void decode_softmax_kernel(
    const float* __restrict__ logits,   // [B*N, 91]
    const float* __restrict__ regs,     // [B*N, 364]
    const float* __restrict__ props,    // [B*N, 4]
    float* __restrict__ cscore,         // [B, CAND]
    int*   __restrict__ cidx,           // [B, CAND]
    float* __restrict__ cbox,           // [B, CAND, 4]
    int*   __restrict__ ccnt)           // [B]
{
  const int wave = (blockIdx.x * blockDim.x + threadIdx.x) >> 5;
  const int lane = threadIdx.x & 31;
  if (wave >= B_IMG * N_PROP) return;
  const int img = wave / N_PROP;
  const int n   = wave - img * N_PROP;

  const float* lrow = logits + (size_t)wave * NCLS;
  // softmax denominator (wave32 shuffle reductions)
  float mx = -INFINITY;
  for (int c = lane; c < NCLS; c += 32) mx = fmaxf(mx, lrow[c]);
#pragma unroll
  for (int s = 16; s; s >>= 1) mx = fmaxf(mx, __shfl_xor(mx, s, 32));
  float den = 0.0f;
  for (int c = lane; c < NCLS; c += 32) den += expf(lrow[c] - mx);
#pragma unroll
  for (int s = 16; s; s >>= 1) den += __shfl_xor(den, s, 32);

  const float* p = props + (size_t)wave * 4;
  const float pw  = p[2] - p[0];
  const float ph  = p[3] - p[1];
  const float pcx = p[0] + 0.5f * pw;
  const float pcy = p[1] + 0.5f * ph;
  const float* rrow = regs + (size_t)wave * (4 * NCLS);

  for (int c = 1 + lane; c < NCLS; c += 32) {
    const float dx = rrow[4 * c + 0] / 10.0f;
    const float dy = rrow[4 * c + 1] / 10.0f;
    const float dw = fminf(rrow[4 * c + 2] / 5.0f, BBOX_CLIP);
    const float dh = fminf(rrow[4 * c + 3] / 5.0f, BBOX_CLIP);
    const float ncx = dx * pw + pcx;
    const float ncy = dy * ph + pcy;
    const float nw  = expf(dw) * pw;
    const float nh  = expf(dh) * ph;
    float x1 = fminf(fmaxf(ncx - 0.5f * nw, 0.0f), IMG_W_F);
    float y1 = fminf(fmaxf(ncy - 0.5f * nh, 0.0f), IMG_H_F);
    float x2 = fminf(fmaxf(ncx + 0.5f * nw, 0.0f), IMG_W_F);
    float y2 = fminf(fmaxf(ncy + 0.5f * nh, 0.0f), IMG_H_F);
    const float sc = expf(lrow[c] - mx) / den;
    if (sc > 0.05f && (x2 - x1) >= 0.01f && (y2 - y1) >= 0.01f) {
      const int slot = atomicAdd(&ccnt[img], 1);
      cscore[(size_t)img * CAND + slot] = sc;
      cidx  [(size_t)img * CAND + slot] = n * CM1 + (c - 1);   // original flat index
      float* bo = cbox + ((size_t)img * CAND + slot) * 4;
      bo[0] = x1; bo[1] = y1; bo[2] = x2; bo[3] = y2;
    }
  }
}

// ---------------- class-aware NMS, one block per image ----------------
__global__ __launch_bounds__(1024) void nms_kernel(
    const float* __restrict__ cbox, float* __restrict__ cscore,
    const int* __restrict__ cidx, const int* __restrict__ ccnt,
    float* __restrict__ out_boxes, float* __restrict__ out_scores,
    int* __restrict__ out_labels)
{
  const int img = blockIdx.x;
  const int tid = threadIdx.x;
  int cnt = ccnt[img];
  if (cnt > CAND) cnt = CAND;
  float*       s  = cscore + (size_t)img * CAND;
  const float* bx = cbox   + (size_t)img * CAND * 4;
  const int*   ix = cidx   + (size_t)img * CAND;

  __shared__ float rS[1024];
  __shared__ int   rI[1024];
  __shared__ int   rO[1024];
  __shared__ float jb[4];
  __shared__ int   jslot, jrem;

  for (int t = 0; t < DETS; ++t) {
    // strided argmax, tie-break on lowest original index (matches jnp.argmax)
    float bs = -INFINITY; int bi = -1, bo = INT_MAX;
    for (int i = tid; i < cnt; i += 1024) {
      const float v = s[i];
      const int oi  = ix[i];
      if (v > bs || (v == bs && oi < bo)) { bs = v; bi = i; bo = oi; }
    }
    rS[tid] = bs; rI[tid] = bi; rO[tid] = bo;
    __syncthreads();
    for (int off = 512; off; off >>= 1) {
      if (tid < off) {
        const float v2 = rS[tid + off];
        const int   o2 = rO[tid + off];
        if (v2 > rS[tid] || (v2 == rS[tid] && o2 < rO[tid])) {
          rS[tid] = v2; rI[tid] = rI[tid + off]; rO[tid] = o2;
        }
      }
      __syncthreads();
    }

    if (tid == 0) {
      const int j = rI[0];
      const bool vj = (j >= 0) && (rS[0] > 0.0f);
      jslot = vj ? j : -1;
      float* ob = out_boxes + ((size_t)img * DETS + t) * 4;
      if (vj) {
        const float* b = bx + (size_t)j * 4;
        jb[0] = b[0]; jb[1] = b[1]; jb[2] = b[2]; jb[3] = b[3];
        jrem = ix[j] % CM1;
        ob[0] = b[0]; ob[1] = b[1]; ob[2] = b[2]; ob[3] = b[3];
        out_scores[img * DETS + t] = rS[0];
        out_labels[img * DETS + t] = jrem + 1;
        s[j] = -1.0f;
      } else {
        ob[0] = 0.f; ob[1] = 0.f; ob[2] = 0.f; ob[3] = 0.f;
        out_scores[img * DETS + t] = 0.0f;
        out_labels[img * DETS + t] = 0;
      }
    }
    __syncthreads();

    if (jslot >= 0) {
      const float jx1 = jb[0], jy1 = jb[1], jx2 = jb[2], jy2 = jb[3];
      const float ja  = (jx2 - jx1) * (jy2 - jy1);
      const int   jr  = jrem;
      for (int i = tid; i < cnt; i += 1024) {
        if (s[i] <= 0.0f) continue;             // already invalid/suppressed
        if ((ix[i] % CM1) != jr) continue;      // offset trick: other classes never overlap
        const float* b = bx + (size_t)i * 4;
        const float lx = fmaxf(jx1, b[0]), ly = fmaxf(jy1, b[1]);
        const float rx = fminf(jx2, b[2]), ry = fminf(jy2, b[3]);
        const float iw = fmaxf(rx - lx, 0.0f), ih = fmaxf(ry - ly, 0.0f);
        const float inter = iw * ih;
        const float ia = (b[2] - b[0]) * (b[3] - b[1]);
        const float iou = inter / (ja + ia - inter);
        if (iou > 0.5f) s[i] = -1.0f;
      }
    }
    __syncthreads();
  }
}

// ---------------- host launch ----------------
extern "C" void kernel_launch(void* const* d_in, const int* in_sizes, int n_in,
                              void* d_out, int out_size, void* d_ws, size_t ws_size,
                              hipStream_t stream) {
  const float* feat  = (const float*)d_in[0];   // [16384, 1024]
  const float* Wc    = (const float*)d_in[1];   // [1024, 91]
  const float* bc    = (const float*)d_in[2];   // [91]
  const float* Wr    = (const float*)d_in[3];   // [1024, 364]
  const float* br    = (const float*)d_in[4];   // [364]
  const float* props = (const float*)d_in[5];   // [4, 4096, 4]

  char* ws = (char*)d_ws;
  size_t off = 0;
  auto alloc = [&](size_t bytes) -> void* {
    void* p = ws + off;
    off = (off + bytes + 255) & ~(size_t)255;
    return p;
  };
  float* logits = (float*)alloc((size_t)(B_IMG * N_PROP) * NCLS * sizeof(float));
  float* regs   = (float*)alloc((size_t)(B_IMG * N_PROP) * 4 * NCLS * sizeof(float));
  float* cscore = (float*)alloc((size_t)B_IMG * CAND * sizeof(float));
  int*   cidx   = (int*)  alloc((size_t)B_IMG * CAND * sizeof(int));
  float* cbox   = (float*)alloc((size_t)B_IMG * CAND * 4 * sizeof(float));
  int*   ccnt   = (int*)  alloc(B_IMG * sizeof(int));

  float* outf       = (float*)d_out;
  float* out_boxes  = outf;                          // [4,100,4]
  float* out_scores = outf + B_IMG * DETS * 4;       // [4,100]
  int*   out_labels = (int*)(outf + B_IMG * DETS * 5);  // [4,100] int32

  zero_cnt_kernel<<<1, 32, 0, stream>>>(ccnt);

  const int M = B_IMG * N_PROP;
  dim3 gcls(M / MT, (NCLS + NT - 1) / NT);
  gemm_bias_wmma<<<gcls, 256, 0, stream>>>(feat, Wc, bc, logits, NCLS);
  dim3 greg(M / MT, (4 * NCLS + NT - 1) / NT);
  gemm_bias_wmma<<<greg, 256, 0, stream>>>(feat, Wr, br, regs, 4 * NCLS);

  decode_softmax_kernel<<<(M * 32) / 256, 256, 0, stream>>>(
      logits, regs, props, cscore, cidx, cbox, ccnt);

  nms_kernel<<<B_IMG, 1024, 0, stream>>>(
      cbox, cscore, cidx, ccnt, out_boxes, out_scores, out_labels);
}